// GINConv2d_81638738362646
// MI455X (gfx1250) — compile-verified
//
#include <hip/hip_runtime.h>

#define C_IN    128
#define C_OUT   128
#define K_NBR   16
#define GROUPS  4
#define CG      32          // channels per group
#define TILE    64          // nodes per workgroup
#define THREADS 256         // 8 waves (wave32)
#define HS      72          // LDS row stride (words): 2*HS % 64 == 16 -> halves on disjoint banks
#define BN_EPS  1e-5f

typedef float v2f __attribute__((ext_vector_type(2)));
typedef float v8f __attribute__((ext_vector_type(8)));

__global__ __launch_bounds__(THREADS)
void gin_fused_kernel(const float* __restrict__ x,
                      const int*   __restrict__ edge,   // (2, N, 16), row 0 used
                      const float* __restrict__ eps,
                      const float* __restrict__ W,      // (128, 32)
                      const float* __restrict__ bias,
                      const float* __restrict__ gamma,
                      const float* __restrict__ beta,
                      const float* __restrict__ mean,
                      const float* __restrict__ var,
                      float*       __restrict__ out,
                      int N)
{
    __shared__ float h_lds[C_IN * HS];          // 36 KB: h[c][node_local]
    __shared__ int   idx_lds[K_NBR * TILE];     // 4 KB, transposed: idx[k][node_local]

    const int t     = threadIdx.x;
    const int node0 = blockIdx.x * TILE;

    // ---- stage neighbor indices into LDS (transposed for conflict-free reads) ----
    for (int i = t; i < TILE * K_NBR; i += THREADS) {
        const int nl = i >> 4;          // local node
        const int k  = i & 15;          // neighbor slot
        const int n  = node0 + nl;
        idx_lds[k * TILE + nl] = (n < N) ? edge[n * K_NBR + k] : 0;
    }
    __syncthreads();

    // ---- gather + self term: h[c][nl] = (1+eps)*x[c][n] + sum_k x[c][idx[n][k]] ----
    {
        const float eps1  = 1.0f + eps[0];
        const int   nl    = t & (TILE - 1);
        const int   cbase = (t >> 6) * CG;      // 4 channel-quarters == the 4 groups
        const int   n     = node0 + nl;

        float acc[CG];
        #pragma unroll
        for (int j = 0; j < CG; ++j) acc[j] = 0.0f;

        if (n < N) {
            for (int k = 0; k < K_NBR; ++k) {
                const int jn = idx_lds[k * TILE + nl];
                const float* xp = x + (size_t)cbase * N + jn;
                #pragma unroll
                for (int j = 0; j < CG; ++j)
                    acc[j] += xp[(size_t)j * N];
            }
            const float* xs = x + (size_t)cbase * N + n;
            #pragma unroll
            for (int j = 0; j < CG; ++j)
                acc[j] = fmaf(eps1, xs[(size_t)j * N], acc[j]);
        }
        #pragma unroll
        for (int j = 0; j < CG; ++j)
            h_lds[(cbase + j) * HS + nl] = acc[j];
    }
    __syncthreads();

    // ---- grouped GEMM via V_WMMA_F32_16X16X4_F32 + fused bias/BN/ReLU ----
    const int wid  = t >> 5;            // wave 0..7
    const int lane = t & 31;
    const int half = lane >> 4;         // which 16-lane half
    const int lr   = lane & 15;

    const int g  = wid >> 1;            // group 0..3
    const int mt = wid & 1;             // 16-row m-tile within the group's 32 outputs
    const int orow = g * CG + mt * 16 + lr;     // A-matrix row = output channel (lanes 0-15 & 16-31 both map M=lr)

    // Preload all A fragments for K=0..31 (step 4).
    // 16x4 f32 A layout: VGPR0 = {K0 | K2}, VGPR1 = {K1 | K3} across the two lane halves.
    v2f a[8];
    #pragma unroll
    for (int ks = 0; ks < 8; ++ks) {
        const int k = ks * 4 + half * 2;
        a[ks].x = W[orow * CG + k + 0];
        a[ks].y = W[orow * CG + k + 1];
    }

    // BN folded constants per accumulator row: D VGPR r holds channel m = mt*16 + r + 8*half.
    float inv[8], sh[8];
    #pragma unroll
    for (int r = 0; r < 8; ++r) {
        const int o  = g * CG + mt * 16 + r + half * 8;
        const float iv = gamma[o] * rsqrtf(var[o] + BN_EPS);
        inv[r] = iv;
        sh[r]  = fmaf(iv, bias[o] - mean[o], beta[o]);
    }

    for (int nt = 0; nt < TILE / 16; ++nt) {
        const int col = nt * 16 + lr;
        v8f acc = {};
        #pragma unroll
        for (int ks = 0; ks < 8; ++ks) {
            const int k = g * CG + ks * 4 + half * 2;   // absolute input-channel row in h_lds
            v2f b;                                      // 4x16 B: VGPR0 = {K0 | K2}, VGPR1 = {K1 | K3}
            b.x = h_lds[(k + 0) * HS + col];
            b.y = h_lds[(k + 1) * HS + col];
            acc = __builtin_amdgcn_wmma_f32_16x16x4_f32(
                      /*neg_a=*/false, a[ks], /*neg_b=*/false, b,
                      /*c_mod=*/(short)0, acc, /*reuse_a=*/false, /*reuse_b=*/false);
        }

        const int n = node0 + nt * 16 + lr;
        if (n < N) {
            #pragma unroll
            for (int r = 0; r < 8; ++r) {
                const int o = g * CG + mt * 16 + r + half * 8;
                const float v = fmaf(inv[r], acc[r], sh[r]);
                out[(size_t)o * N + n] = fmaxf(v, 0.0f);
            }
        }
    }
}

extern "C" void kernel_launch(void* const* d_in, const int* in_sizes, int n_in,
                              void* d_out, int out_size, void* d_ws, size_t ws_size,
                              hipStream_t stream) {
    const float* x      = (const float*)d_in[0];
    const int*   edge   = (const int*)  d_in[1];
    const float* eps    = (const float*)d_in[2];
    const float* conv_w = (const float*)d_in[3];
    const float* conv_b = (const float*)d_in[4];
    const float* gamma  = (const float*)d_in[5];
    const float* beta   = (const float*)d_in[6];
    const float* mean   = (const float*)d_in[7];
    const float* var    = (const float*)d_in[8];
    float* out = (float*)d_out;

    const int N = in_sizes[0] / C_IN;   // 50000
    const int blocks = (N + TILE - 1) / TILE;
    gin_fused_kernel<<<blocks, THREADS, 0, stream>>>(
        x, edge, eps, conv_w, conv_b, gamma, beta, mean, var, out, N);
}